// PennyLaneQuantumLayer_58110907515676
// MI455X (gfx1250) — compile-verified
//
#include <hip/hip_runtime.h>
#include <hip/hip_bf16.h>

typedef __attribute__((ext_vector_type(2))) float v2f;
typedef __attribute__((ext_vector_type(8))) float v8f;

#define N_BATCH  2048
#define N_QUBITS 10
#define DIM      1024
#define Q_LAYERS 3

// workspace layout (float offsets)
#define OFF_GATES 0u
#define OFF_C     256u
#define OFF_S     (OFF_C + 20480u)
#define OFF_WRE   (OFF_S + 20480u)
#define OFF_WIM   (OFF_WRE + 1048576u)
#define OFF_B     (OFF_WIM + 1048576u)
#define OFF_PART  (OFF_B + 2097152u)   // 128 x 2048 partials

// ---------------------------------------------------------------------------
// 1) Build the 30 Rot(phi,theta,omega) 2x2 complex gate matrices.
__global__ void build_gates(const float* __restrict__ w, float* __restrict__ gates) {
    int g = threadIdx.x;
    if (g >= Q_LAYERS * N_QUBITS) return;
    float phi = w[g * 3 + 0], th = w[g * 3 + 1], om = w[g * 3 + 2];
    float ct = cosf(0.5f * th), st = sinf(0.5f * th);
    float cp = cosf(0.5f * (phi + om)), sp = sinf(0.5f * (phi + om));
    float cm = cosf(0.5f * (phi - om)), sm = sinf(0.5f * (phi - om));
    float* o = gates + g * 8;
    o[0] =  cp * ct; o[1] = -sp * ct;   // U00 = e^{-i(phi+om)/2} ct
    o[2] = -cm * st; o[3] = -sm * st;   // U01 = -e^{+i(phi-om)/2} st
    o[4] =  cm * st; o[5] = -sm * st;   // U10 = e^{-i(phi-om)/2} st
    o[6] =  cp * ct; o[7] =  sp * ct;   // U11 = e^{+i(phi+om)/2} ct
}

// ---------------------------------------------------------------------------
// 2) cos/sin(x/2) tables.
__global__ void build_cs(const float* __restrict__ x, float* __restrict__ c,
                         float* __restrict__ s) {
    int i = blockIdx.x * blockDim.x + threadIdx.x;
    if (i < N_BATCH * N_QUBITS) {
        float h = 0.5f * x[i];
        c[i] = cosf(h);
        s[i] = sinf(h);
    }
}

// ---------------------------------------------------------------------------
// 3) B[j][n] = Psi[n][j] = prod_i (bit_{9-i}(j) ? s[n,i] : c[n,i]).
//    Layout [j][n] (n fastest) so GEMM B-fragment loads are lane-contiguous.
__global__ __launch_bounds__(256) void build_B(const float* __restrict__ c,
                                               const float* __restrict__ s,
                                               float* __restrict__ B) {
    int j = blockIdx.x;
    for (int n = threadIdx.x; n < N_BATCH; n += 256) {
        const float* cr = c + n * N_QUBITS;
        const float* sr = s + n * N_QUBITS;
        float p = 1.0f;
#pragma unroll
        for (int i = 0; i < N_QUBITS; ++i)
            p *= ((j >> (N_QUBITS - 1 - i)) & 1) ? sr[i] : cr[i];
        B[(size_t)j * N_BATCH + n] = p;
    }
}

// ---------------------------------------------------------------------------
// 4) Build W (the fixed 3-layer circuit) column by column: block j simulates
//    the circuit on basis state e_j with the 1024-amp statevector in LDS.
__global__ __launch_bounds__(512) void build_W(const float* __restrict__ gates,
                                               float* __restrict__ Wre,
                                               float* __restrict__ Wim) {
    __shared__ float sre[DIM];
    __shared__ float sim_[DIM];
    const int t = threadIdx.x;      // 0..511
    const int j = blockIdx.x;       // basis column
    for (int idx = t; idx < DIM; idx += 512) {
        sre[idx] = (idx == j) ? 1.0f : 0.0f;
        sim_[idx] = 0.0f;
    }
    __syncthreads();

    for (int l = 0; l < Q_LAYERS; ++l) {
        // Rot gates on each wire
        for (int i = 0; i < N_QUBITS; ++i) {
            const float* g = gates + (l * N_QUBITS + i) * 8;
            float u00r = g[0], u00i = g[1], u01r = g[2], u01i = g[3];
            float u10r = g[4], u10i = g[5], u11r = g[6], u11i = g[7];
            int p   = N_QUBITS - 1 - i;                 // bit position of wire i
            int low = t & ((1 << p) - 1);
            int i0  = ((t >> p) << (p + 1)) | low;      // pair with bit p = 0
            int i1  = i0 | (1 << p);
            float a0r = sre[i0], a0i = sim_[i0];
            float a1r = sre[i1], a1i = sim_[i1];
            sre[i0]  = u00r * a0r - u00i * a0i + u01r * a1r - u01i * a1i;
            sim_[i0] = u00r * a0i + u00i * a0r + u01r * a1i + u01i * a1r;
            sre[i1]  = u10r * a0r - u10i * a0i + u11r * a1r - u11i * a1i;
            sim_[i1] = u10r * a0i + u10i * a0r + u11r * a1i + u11i * a1r;
            __syncthreads();
        }
        // CNOT ring
        for (int i = 0; i < N_QUBITS; ++i) {
            int pc = N_QUBITS - 1 - i;                     // control bit pos
            int pt = N_QUBITS - 1 - ((i + 1) % N_QUBITS);  // target bit pos
            if (t < 256) {
                int ps = (pc < pt) ? pc : pt;
                int pb = (pc < pt) ? pt : pc;
                int bs = (ps == pc) ? 1 : 0;   // control=1, target=0
                int bb = (pb == pc) ? 1 : 0;
                int m  = t;
                int lo = m & ((1 << ps) - 1);
                m = ((m >> ps) << (ps + 1)) | lo | (bs << ps);
                lo = m & ((1 << pb) - 1);
                m = ((m >> pb) << (pb + 1)) | lo | (bb << pb);
                int m2 = m | (1 << pt);        // target bit set
                float tr = sre[m];  sre[m]  = sre[m2];  sre[m2]  = tr;
                float ti = sim_[m]; sim_[m] = sim_[m2]; sim_[m2] = ti;
            }
            __syncthreads();
        }
    }
    for (int idx = t; idx < DIM; idx += 512) {
        Wre[(size_t)idx * DIM + j] = sre[idx];
        Wim[(size_t)idx * DIM + j] = sim_[idx];
    }
}

// ---------------------------------------------------------------------------
// 5) WMMA GEMM: Phi = W * Psi^T (re & im), fused sign-weighted |.|^2 epilogue.
//    One wave per 16x16 output tile; K = 1024 in steps of 4 via
//    v_wmma_f32_16x16x4_f32. 8 waves/block share one N-tile (B reuse in L0).
__global__ __launch_bounds__(256) void gemm_expval(const float* __restrict__ Wre,
                                                   const float* __restrict__ Wim,
                                                   const float* __restrict__ B,
                                                   float* __restrict__ part) {
    const int wave    = threadIdx.x >> 5;
    const int lane    = threadIdx.x & 31;
    const int lane_lo = lane & 15;
    const int lane_hi = lane >> 4;
    const int Ntile   = blockIdx.x & 127;                 // 128 N tiles
    const int Mtile   = ((blockIdx.x >> 7) << 3) + wave;  // 64 M tiles
    const int Mbase   = Mtile * 16;
    const int Nbase   = Ntile * 16;

    // A fragment: lane (lane_hi*16+lane_lo) holds A[Mbase+lane_lo, k0+2*lane_hi + {0,1}]
    const float* aRe = Wre + (size_t)(Mbase + lane_lo) * DIM + 2 * lane_hi;
    const float* aIm = Wim + (size_t)(Mbase + lane_lo) * DIM + 2 * lane_hi;
    // B fragment: lane holds B[k0+2*lane_hi + {0,1}, Nbase+lane_lo]
    const float* bP  = B + (size_t)(2 * lane_hi) * N_BATCH + Nbase + lane_lo;

    v8f accRe = {};
    v8f accIm = {};
    for (int k0 = 0; k0 < DIM; k0 += 4) {
        v2f a_re = *(const v2f*)(aRe + k0);
        v2f a_im = *(const v2f*)(aIm + k0);
        v2f b;
        b.x = bP[(size_t)k0 * N_BATCH];
        b.y = bP[(size_t)(k0 + 1) * N_BATCH];
        accRe = __builtin_amdgcn_wmma_f32_16x16x4_f32(false, a_re, false, b,
                                                      (short)0, accRe, false, false);
        accIm = __builtin_amdgcn_wmma_f32_16x16x4_f32(false, a_im, false, b,
                                                      (short)0, accIm, false, false);
    }

    // C/D layout: lane_lo = N, vgpr j holds M = Mbase + j + 8*lane_hi.
    float ssum = 0.0f;
#pragma unroll
    for (int jj = 0; jj < 8; ++jj) {
        int   m  = Mbase + jj + 8 * lane_hi;
        float sg = (__popc(m) & 1) ? -1.0f : 1.0f;
        ssum += sg * (accRe[jj] * accRe[jj] + accIm[jj] * accIm[jj]);
    }
    part[(size_t)(Mtile * 2 + lane_hi) * N_BATCH + Nbase + lane_lo] = ssum;
}

// ---------------------------------------------------------------------------
// 6) Deterministic reduction over the 128 M-partials.
__global__ void reduce_out(const float* __restrict__ part, float* __restrict__ out) {
    int n = blockIdx.x * blockDim.x + threadIdx.x;
    if (n < N_BATCH) {
        float s = 0.0f;
        for (int r = 0; r < 128; ++r) s += part[(size_t)r * N_BATCH + n];
        out[n] = s;
    }
}

// ---------------------------------------------------------------------------
extern "C" void kernel_launch(void* const* d_in, const int* in_sizes, int n_in,
                              void* d_out, int out_size, void* d_ws, size_t ws_size,
                              hipStream_t stream) {
    const float* x = (const float*)d_in[0];   // (2048, 10)
    const float* w = (const float*)d_in[1];   // (3, 10, 3)
    float* ws    = (float*)d_ws;
    float* gates = ws + OFF_GATES;
    float* c     = ws + OFF_C;
    float* s     = ws + OFF_S;
    float* Wre   = ws + OFF_WRE;
    float* Wim   = ws + OFF_WIM;
    float* B     = ws + OFF_B;
    float* part  = ws + OFF_PART;
    float* out   = (float*)d_out;

    build_gates<<<1, 32, 0, stream>>>(w, gates);
    build_cs<<<(N_BATCH * N_QUBITS + 255) / 256, 256, 0, stream>>>(x, c, s);
    build_B<<<DIM, 256, 0, stream>>>(c, s, B);
    build_W<<<DIM, 512, 0, stream>>>(gates, Wre, Wim);
    gemm_expval<<<1024, 256, 0, stream>>>(Wre, Wim, B, part);
    reduce_out<<<(N_BATCH + 255) / 256, 256, 0, stream>>>(part, out);
}